// TinyTransducer_87917980549125
// MI455X (gfx1250) — compile-verified
//
#include <hip/hip_runtime.h>

typedef float v2f __attribute__((ext_vector_type(2)));
typedef float v4f __attribute__((ext_vector_type(4)));
typedef float v8f __attribute__((ext_vector_type(8)));

// Y = X * W^T
//   X: M x K row-major, row stride ldx
//   W: N x K row-major, row stride ldw (lets us address the enc/dec halves of fc_weight)
//   Y: M x N row-major, row stride ldy
// Grid: (N/64, M/16); block: 128 threads = 4 wave32, each wave owns one 16x16 tile.
// Uses V_WMMA_F32_16X16X4_F32; all tile dims divide evenly so EXEC is all-ones.
__global__ __launch_bounds__(128)
void tt_gemm_nt_f32_wmma(const float* __restrict__ X, const float* __restrict__ W,
                         float* __restrict__ Y, int K, int ldx, int ldw, int ldy)
{
    const int lane = threadIdx.x & 31;
    const int wave = threadIdx.x >> 5;      // 0..3 -> 16-wide column slice
    const int half = lane >> 4;             // 0: K={0,1} lanes, 1: K={2,3} lanes
    const int r16  = lane & 15;

    const int row0 = blockIdx.y * 16;
    const int col0 = blockIdx.x * 64 + wave * 16;

    // A 16x4 fragment: lane holds X[row0+r16][k + 2*half .. +1]   (v2f per k-step)
    // B 4x16 fragment: lane holds W[col0+r16][k + 2*half .. +1]
    const float* __restrict__ a = X + (size_t)(row0 + r16) * (size_t)ldx + 2 * half;
    const float* __restrict__ b = W + (size_t)(col0 + r16) * (size_t)ldw + 2 * half;

    v8f acc = {};
#pragma unroll 8
    for (int k = 0; k < K; k += 4) {
        v2f av = *(const v2f*)(a + k);
        v2f bv = *(const v2f*)(b + k);
        // 8 args: (neg_a, A, neg_b, B, c_mod, C, reuse_a, reuse_b)
        acc = __builtin_amdgcn_wmma_f32_16x16x4_f32(
            false, av, false, bv, (short)0, acc, false, false);
    }

    // C/D layout: VGPR r, lanes 0-15 -> row r; lanes 16-31 -> row r+8
    float* __restrict__ y = Y + (size_t)(row0 + 8 * half) * (size_t)ldy + col0 + r16;
#pragma unroll
    for (int r = 0; r < 8; ++r)
        y[(size_t)r * (size_t)ldy] = acc[r];
}

// out[b,t,u,v] = enc_proj[b,t,v] + dec_proj[b,u,v]
// One block per (b,t). 128 threads cover V=512 as float4. enc row pinned in
// registers; dec rows (0.75 MB total) stay in L2 across the u loop. The 400 MB
// output stream uses non-temporal b128 stores.
__global__ __launch_bounds__(128)
void tt_bcast_add(const float* __restrict__ encp, const float* __restrict__ decp,
                  float* __restrict__ out, int T, int U, int V)
{
    const int bt = blockIdx.x;          // b*T + t
    const int b  = bt / T;
    const int v  = threadIdx.x * 4;     // 0..508

    const v4f e = *(const v4f*)(encp + (size_t)bt * (size_t)V + v);
    const float* __restrict__ dp = decp + (size_t)b * (size_t)U * (size_t)V + v;
    float* __restrict__ op = out + (size_t)bt * (size_t)U * (size_t)V + v;

#pragma unroll 4
    for (int u = 0; u < U; ++u) {
        v4f d = *(const v4f*)(dp + (size_t)u * (size_t)V);
        v4f s = e + d;
        __builtin_nontemporal_store(s, (v4f*)(op + (size_t)u * (size_t)V));
    }
}

extern "C" void kernel_launch(void* const* d_in, const int* in_sizes, int n_in,
                              void* d_out, int out_size, void* d_ws, size_t ws_size,
                              hipStream_t stream)
{
    (void)in_sizes; (void)n_in; (void)out_size; (void)ws_size;

    constexpr int B = 4, T = 512, U = 96, D = 512, V = 512;

    const float* enc = (const float*)d_in[0];   // [B,T,D]
    const float* dec = (const float*)d_in[1];   // [B,U,D]
    const float* fcw = (const float*)d_in[2];   // [V, 2D] row-major
    float* out = (float*)d_out;                 // [B,T,U,V]

    // Workspace: enc_proj [B*T,V] then dec_proj [B*U,V]  (~5 MB total)
    float* enc_proj = (float*)d_ws;
    float* dec_proj = enc_proj + (size_t)B * T * V;

    dim3 blk(128);

    // enc_proj = enc (2048 x 512) * W_enc^T   (W_enc = fcw[:, :D], row stride 2D)
    dim3 g_enc(V / 64, (B * T) / 16);
    tt_gemm_nt_f32_wmma<<<g_enc, blk, 0, stream>>>(enc, fcw, enc_proj,
                                                   D, D, 2 * D, V);

    // dec_proj = dec (384 x 512) * W_dec^T    (W_dec = fcw[:, D:], row stride 2D)
    dim3 g_dec(V / 64, (B * U) / 16);
    tt_gemm_nt_f32_wmma<<<g_dec, blk, 0, stream>>>(dec, fcw + D, dec_proj,
                                                   D, D, 2 * D, V);

    // out = enc_proj[:, :, None, :] + dec_proj[:, None, :, :]
    tt_bcast_add<<<B * T, blk, 0, stream>>>(enc_proj, dec_proj, out, T, U, V);
}